// costVolComplex_77464030150981
// MI455X (gfx1250) — compile-verified
//
#include <hip/hip_runtime.h>

// Problem constants
#define HH 96
#define WW 80
#define DD 128
#define NV (HH * WW * DD)          // 983040 voxels
#define NOFF 27

// 1D Gaussian (sigma=0.5, 3 taps): exp(-2)=0.13533528..., normalized
#define WCEN 0.78698604216159846f
#define WSIDE 0.10650697891920077f

typedef __attribute__((ext_vector_type(2))) float v2f;
typedef __attribute__((ext_vector_type(4))) float v4f;
typedef __attribute__((ext_vector_type(8))) float v8f;

// ---------------------------------------------------------------------------
// Per-lane A-matrix (blur weight) table, computed at compile time.
// Lane L (hi = L>>4, nn = L&15) needs, for WMMA step s: wgt(nn, 4s+2*hi) and
// wgt(nn, 4s+2*hi+1).  8 floats per lane, 32B aligned -> two b128 loads.
// ---------------------------------------------------------------------------
constexpr float wgt_c(int m, int k) {
    int d = k - m;
    return (d == 0) ? WCEN : ((d == 1 || d == -1) ? WSIDE : 0.0f);
}

struct WTab { float v[256]; };

constexpr WTab make_wtab() {
    WTab t{};
    for (int lane = 0; lane < 32; ++lane) {
        int hi = lane >> 4, nn = lane & 15;
        for (int s = 0; s < 4; ++s) {
            int k0 = 4 * s + 2 * hi;
            t.v[lane * 8 + 2 * s + 0] = wgt_c(nn, k0);
            t.v[lane * 8 + 2 * s + 1] = wgt_c(nn, k0 + 1);
        }
    }
    return t;
}

__device__ __constant__ WTab g_wtab = make_wtab();

// ---------------------------------------------------------------------------
// Kernel 1: cost volume  cv[o][z][y][x] = -| src[z+dz][y+dy][x+dx] - tgt[z][y][x] |
// (zero padding on src, matching jnp.pad)
// ---------------------------------------------------------------------------
__global__ __launch_bounds__(256) void costvol_kernel(
    const float* __restrict__ src, const float* __restrict__ tgt,
    float* __restrict__ cv)
{
    int v = blockIdx.x * blockDim.x + threadIdx.x;
    if (v >= NV) return;
    int x = v % DD;
    int y = (v / DD) % WW;
    int z = v / (DD * WW);
    float t = tgt[v];
    int o = 0;
    #pragma unroll
    for (int i = -1; i <= 1; ++i)
      #pragma unroll
      for (int j = -1; j <= 1; ++j)
        #pragma unroll
        for (int k = -1; k <= 1; ++k) {
            int zz = z + i, yy = y + j, xx = x + k;
            float s = 0.0f;
            if ((unsigned)zz < HH && (unsigned)yy < WW && (unsigned)xx < DD)
                s = src[(zz * WW + yy) * DD + xx];
            cv[o * NV + v] = -fabsf(s - t);
            ++o;
        }
}

// ---------------------------------------------------------------------------
// Kernel 2: 1D 3-tap Gaussian blur along a compile-time axis, via
// V_WMMA_F32_16X16X4_F32.  One wave32 handles a 16(axis) x 16(cols) tile:
//   D(16x16) = W(16x16, tridiagonal) * X(16x16)  as 4 chained K=4 WMMAs,
// with the two out-of-tile halo taps folded into the C accumulator.
// VGPR layouts per CDNA5 ISA 7.12.2:
//   A(16x4):  lane<16 -> M=lane, K={0,1}; lane>=16 -> M=lane-16, K={2,3}
//   B(4x16):  lane<16 -> N=lane, K={0,1}; lane>=16 -> N=lane-16, K={2,3}
//   D(16x16): VGPR r -> M=r (lane<16, N=lane) / M=r+8 (lane>=16, N=lane-16)
// SA==1 specialization: B pairs are 8B-contiguous (b64 loads) and each lane's
// 8 outputs are 16B-aligned contiguous (2x b128 stores).
// ---------------------------------------------------------------------------
template <int SA, int LA, int SC, int LC, int NOB, int SOB>
__global__ __launch_bounds__(256) void blur1d_wmma_kernel(
    const float* __restrict__ in, float* __restrict__ out, int n_ch)
{
    constexpr int nA = LA / 16;
    constexpr int nC = LC / 16;
    int wid = blockIdx.x * 8 + (threadIdx.x >> 5);
    int total = n_ch * NOB * nA * nC;
    if (wid >= total) return;   // uniform per wave: EXEC stays all-ones for WMMA

    int ta = wid % nA;  int t = wid / nA;
    int tc = t % nC;    t /= nC;
    int oB = t % NOB;
    int oA = t / NOB;

    int axis0 = ta << 4;
    int col0  = tc << 4;

    int lane = threadIdx.x & 31;
    int hi   = lane >> 4;        // 0: lanes 0-15, 1: lanes 16-31
    int nn   = lane & 15;        // M (for A) / N (for B and D)

    // Per-lane constant weights: two b128 loads from the 1KB constant table
    const float* wp = g_wtab.v + lane * 8;
    v4f w0 = ((const v4f*)wp)[0];
    v4f w1 = ((const v4f*)wp)[1];
    const v2f* wpair0 = (const v2f*)&w0;   // steps 0,1
    const v2f* wpair1 = (const v2f*)&w1;   // steps 2,3

    long long base = (long long)oA * NV + oB * SOB + (col0 + nn) * SC;
    const float* __restrict__ pin  = in  + base;
    float* __restrict__       pout = out + base;

    // Halo taps into the accumulator (out[0] += ws*in[-1], out[15] += ws*in[16])
    v8f c = {};
    if (!hi) {
        if (axis0 > 0)
            c[0] = WSIDE * pin[(axis0 - 1) * SA];
    } else {
        if (axis0 + 16 < LA)
            c[7] = WSIDE * pin[(axis0 + 16) * SA];
    }

    #pragma unroll
    for (int s = 0; s < 4; ++s) {
        int k0 = 4 * s + (hi ? 2 : 0);   // this lane-half's K pair within the tile
        v2f b;
        if constexpr (SA == 1) {
            b = *(const v2f*)(pin + axis0 + k0);          // 8B-aligned b64 load
        } else {
            b[0] = pin[(axis0 + k0) * SA];
            b[1] = pin[(axis0 + k0 + 1) * SA];
        }
        v2f a = (s < 2) ? wpair0[s] : wpair1[s - 2];      // register sub-extract
        c = __builtin_amdgcn_wmma_f32_16x16x4_f32(
                /*neg_a=*/false, a, /*neg_b=*/false, b,
                /*c_mod=*/(short)0, c, /*reuse_a=*/false, /*reuse_b=*/false);
    }

    int m0 = hi ? 8 : 0;
    if constexpr (SA == 1) {
        // 8 contiguous outputs per lane, 16B aligned -> two b128 stores
        *(v4f*)(pout + axis0 + m0)     = *((const v4f*)&c);
        *(v4f*)(pout + axis0 + m0 + 4) = *(((const v4f*)&c) + 1);
    } else {
        #pragma unroll
        for (int r = 0; r < 8; ++r)
            pout[(axis0 + m0 + r) * SA] = c[r];
    }
}

// ---------------------------------------------------------------------------
// Kernel 3: argmax-onehot over 27 channels (optionally flow-coupled) -> flow
// Matches p == p.max tie semantics: sum offsets of all exact-max channels.
// ---------------------------------------------------------------------------
__global__ __launch_bounds__(256) void argmax_flow_kernel(
    const float* __restrict__ cv, const float* __restrict__ flow_in,
    float coeff, int use_flow, float* __restrict__ flow_out)
{
    int v = blockIdx.x * blockDim.x + threadIdx.x;
    if (v >= NV) return;

    float fz = 0.0f, fy = 0.0f, fx = 0.0f;
    if (use_flow) {
        fz = flow_in[v];
        fy = flow_in[NV + v];
        fx = flow_in[2 * NV + v];
    }

    float vals[NOFF];
    float best = -3.4e38f;
    #pragma unroll
    for (int o = 0; o < NOFF; ++o) {
        float c = cv[o * NV + v];
        if (use_flow) {
            float oz = (float)(o / 9 - 1);
            float oy = (float)((o / 3) % 3 - 1);
            float ox = (float)(o % 3 - 1);
            float dz = fz - oz, dy = fy - oy, dx = fx - ox;
            c -= coeff * (dz * dz + dy * dy + dx * dx);
        }
        vals[o] = c;
        best = fmaxf(best, c);
    }

    float sz = 0.0f, sy = 0.0f, sx = 0.0f;
    #pragma unroll
    for (int o = 0; o < NOFF; ++o) {
        if (vals[o] == best) {
            sz += (float)(o / 9 - 1);
            sy += (float)((o / 3) % 3 - 1);
            sx += (float)(o % 3 - 1);
        }
    }
    flow_out[v] = sz;
    flow_out[NV + v] = sy;
    flow_out[2 * NV + v] = sx;
}

// ---------------------------------------------------------------------------
// Host-side orchestration
// ---------------------------------------------------------------------------
static inline void swapf(float*& a, float*& b) { float* t = a; a = b; b = t; }

// Three iterations of the separable 3x3x3 Gaussian (X, Y, Z per iteration).
// Every pass is exactly C*3840 waves -> C*480 blocks of 256 threads.
static void blur3x(float*& cur, float*& tmp, int C, hipStream_t stream)
{
    const int blocks = C * 480;
    for (int it = 0; it < 3; ++it) {
        // X axis (contiguous): axis stride 1/len 128, cols = y (stride 128/len 80)
        blur1d_wmma_kernel<1, DD, DD, WW, HH, WW * DD>
            <<<blocks, 256, 0, stream>>>(cur, tmp, C);
        swapf(cur, tmp);
        // Y axis: axis stride 128/len 80, cols = x (stride 1/len 128)
        blur1d_wmma_kernel<DD, WW, 1, DD, HH, WW * DD>
            <<<blocks, 256, 0, stream>>>(cur, tmp, C);
        swapf(cur, tmp);
        // Z axis: axis stride 10240/len 96, cols = x, outer = (C, y)
        blur1d_wmma_kernel<WW * DD, HH, 1, DD, WW, DD>
            <<<blocks, 256, 0, stream>>>(cur, tmp, C);
        swapf(cur, tmp);
    }
}

extern "C" void kernel_launch(void* const* d_in, const int* in_sizes, int n_in,
                              void* d_out, int out_size, void* d_ws, size_t ws_size,
                              hipStream_t stream)
{
    const float* src = (const float*)d_in[0];
    const float* tgt = (const float*)d_in[1];
    float* out = (float*)d_out;                 // final flow (3 x NV), also flow ping buffer

    float* cvA   = (float*)d_ws;                // 27*NV floats
    float* cvB   = cvA + (long long)NOFF * NV;  // 27*NV floats
    float* flowA = cvB + (long long)NOFF * NV;  // 3*NV floats

    const int vb = NV / 256;                    // 3840 blocks, exact

    // 1) cost volume
    costvol_kernel<<<vb, 256, 0, stream>>>(src, tgt, cvA);

    // 2) blur cv 3x (depthwise over 27 channels); 9 passes, ends in cvB
    float* cv  = cvA;
    float* cvt = cvB;
    blur3x(cv, cvt, NOFF, stream);

    // 3) initial argmax -> flow, then blur 3x (9 passes: result lands in d_out)
    argmax_flow_kernel<<<vb, 256, 0, stream>>>(cv, out, 0.0f, 0, flowA);
    {
        float* f  = flowA;
        float* ft = out;
        blur3x(f, ft, 3, stream);
    }

    // 4) six coupled rounds
    const float coefs[6] = {0.003f, 0.01f, 0.03f, 0.1f, 0.3f, 1.0f};
    for (int r = 0; r < 6; ++r) {
        argmax_flow_kernel<<<vb, 256, 0, stream>>>(cv, out, coefs[r], 1, flowA);
        float* f  = flowA;
        float* ft = out;
        blur3x(f, ft, 3, stream);
    }
}